// TripletLoss_WeightedNorm_66632122630662
// MI455X (gfx1250) — compile-verified
//
#include <hip/hip_runtime.h>
#include <hip/hip_bf16.h>

typedef __attribute__((ext_vector_type(2))) float v2f;
typedef __attribute__((ext_vector_type(8))) float v8f;
typedef __attribute__((ext_vector_type(4))) int   v4i;

#define N_SAMP 1024
#define DFEAT  2048
#define KDIM   16384        // S*D = 8*2048
#define MARGIN 0.3f
#define EPS_D  1e-12f

#define BM  128
#define BN  128
#define KB  32
#define LDK 36              // padded LDS row stride: float4-aligned stores, conflict-free frag reads
#define TILE_F (BM * LDK)   // floats per (buffer, matrix)

// ---- CDNA5 async global->LDS copy (bypasses VGPRs, tracked by ASYNCcnt) ----
#if __has_builtin(__builtin_amdgcn_global_load_async_to_lds_b128)
#define ASYNC_LDS 1
typedef __attribute__((address_space(1))) v4i gas_v4i;   // global (AS1) int4
typedef __attribute__((address_space(3))) v4i las_v4i;   // LDS (AS3) int4
__device__ __forceinline__ void async_copy16(const float* gsrc, float* ldst) {
    __builtin_amdgcn_global_load_async_to_lds_b128(
        (gas_v4i*)(uintptr_t)gsrc,
        (las_v4i*)(uint32_t)(uintptr_t)ldst,  // low 32 bits of generic LDS ptr = LDS offset
        0, 0);
}
__device__ __forceinline__ void wait_async0() {
#if __has_builtin(__builtin_amdgcn_s_wait_asynccnt)
    __builtin_amdgcn_s_wait_asynccnt(0);
#else
    asm volatile("s_wait_asynccnt 0x0" ::: "memory");
#endif
}
#else
#define ASYNC_LDS 0
#endif

// ---------------------------------------------------------------------------
// Kernel 1: sq[i] = sum_k w[k % D] * x[i,k]^2   ( == ||x_i * sqrt(w)||^2 )
// ---------------------------------------------------------------------------
__global__ void tl_sqnorm(const float* __restrict__ x, const float* __restrict__ w,
                          float* __restrict__ sq) {
    __shared__ float red[256];
    const int i = blockIdx.x;
    const float* row = x + (size_t)i * KDIM;
    float acc = 0.f;
    for (int k = threadIdx.x; k < KDIM; k += 256) {
        float v = row[k];
        acc = fmaf(w[k & (DFEAT - 1)] * v, v, acc);
    }
    red[threadIdx.x] = acc;
    __syncthreads();
    for (int s = 128; s > 0; s >>= 1) {
        if (threadIdx.x < s) red[threadIdx.x] += red[threadIdx.x + s];
        __syncthreads();
    }
    if (threadIdx.x == 0) sq[i] = red[0];
}

// ---------------------------------------------------------------------------
// Kernel 2: gram[i,j] = sum_k w[k%D] * x[i,k] * x[j,k]  via V_WMMA_F32_16X16X4_F32
//   256 thr / 8 waves, 128x128 tile, wave owns 64x32 (4x2 subtiles).
//   Double-buffered LDS; A slab prefetched with async global->LDS copy,
//   B slab register-staged + scaled by w (w = sqrt(w)^2 folded on one side).
// ---------------------------------------------------------------------------
__global__ void tl_gram_wmma(const float* __restrict__ x, const float* __restrict__ w,
                             float* __restrict__ gram) {
    __shared__ float As[2 * TILE_F];
    __shared__ float Bs[2 * TILE_F];

    const int tid  = threadIdx.x;
    const int lane = tid & 31;
    const int wave = tid >> 5;
    const int wm   = (wave & 1) * 64;   // wave row offset inside 128-tile
    const int wn   = (wave >> 1) * 32;  // wave col offset inside 128-tile
    const int lr   = lane & 15;         // row (A) / col (B,D) within 16x16 tile
    const int lh   = lane >> 4;         // lane half -> k = 2*lh + v

    const int rowBase = blockIdx.y * BM;
    const int colBase = blockIdx.x * BN;

    // fill mapping: thread -> kc = (tid&7)*4, rows r0 + 32*q
    const int kc = (tid & 7) * 4;
    const int r0 = tid >> 3;

    v8f acc[4][2] = {};

    // ---- slab fill helpers -------------------------------------------------
    auto fillA = [&](int buf, int kb) {
        float* dst = &As[buf * TILE_F];
#if ASYNC_LDS
#pragma unroll
        for (int q = 0; q < 4; ++q) {
            const int r = r0 + q * 32;
            async_copy16(x + (size_t)(rowBase + r) * KDIM + kb + kc, &dst[r * LDK + kc]);
        }
#else
        float4 a[4];
#pragma unroll
        for (int q = 0; q < 4; ++q)
            a[q] = *reinterpret_cast<const float4*>(x + (size_t)(rowBase + r0 + q * 32) * KDIM + kb + kc);
#pragma unroll
        for (int q = 0; q < 4; ++q)
            *reinterpret_cast<float4*>(&dst[(r0 + q * 32) * LDK + kc]) = a[q];
#endif
    };
    auto fillB = [&](int buf, int kb) {
        float* dst = &Bs[buf * TILE_F];
        const int gk = kb + kc;
        const int d0 = gk & (DFEAT - 1);          // gk multiple of 4 -> d0+3 <= 2047
        const float w0 = w[d0], w1 = w[d0 + 1], w2 = w[d0 + 2], w3 = w[d0 + 3];
        float4 b[4];
#pragma unroll
        for (int q = 0; q < 4; ++q)
            b[q] = *reinterpret_cast<const float4*>(x + (size_t)(colBase + r0 + q * 32) * KDIM + gk);
#pragma unroll
        for (int q = 0; q < 4; ++q) {
            b[q].x *= w0; b[q].y *= w1; b[q].z *= w2; b[q].w *= w3;
            *reinterpret_cast<float4*>(&dst[(r0 + q * 32) * LDK + kc]) = b[q];
        }
    };

    // ---- prologue: stage slab 0 -------------------------------------------
    fillA(0, 0);
    fillB(0, 0);
#if ASYNC_LDS
    wait_async0();
#endif
    __syncthreads();   // s_wait_dscnt 0 + barrier

    // ---- main loop: compute slab `cur` while prefetching slab `nxt` -------
    for (int kb = 0; kb < KDIM; kb += KB) {
        const int cur = (kb / KB) & 1;
        const int nxt = cur ^ 1;
        if (kb + KB < KDIM) {
            fillA(nxt, kb + KB);
            fillB(nxt, kb + KB);
        }

        const float* Ab = &As[cur * TILE_F];
        const float* Bb = &Bs[cur * TILE_F];
#pragma unroll
        for (int kk = 0; kk < 8; ++kk) {
            const int kf = kk * 4 + lh * 2;
            v2f bfrag0 = *reinterpret_cast<const v2f*>(&Bb[(wn +  0 + lr) * LDK + kf]);
            v2f bfrag1 = *reinterpret_cast<const v2f*>(&Bb[(wn + 16 + lr) * LDK + kf]);
#pragma unroll
            for (int m = 0; m < 4; ++m) {
                v2f afrag = *reinterpret_cast<const v2f*>(&Ab[(wm + m * 16 + lr) * LDK + kf]);
                acc[m][0] = __builtin_amdgcn_wmma_f32_16x16x4_f32(
                    false, afrag, false, bfrag0, (short)0, acc[m][0], false, false);
                acc[m][1] = __builtin_amdgcn_wmma_f32_16x16x4_f32(
                    false, afrag, false, bfrag1, (short)0, acc[m][1], false, false);
            }
        }
#if ASYNC_LDS
        wait_async0();         // next-slab async A copies landed in LDS
#endif
        __syncthreads();       // next-slab B ds_stores visible + all waves done reading cur
    }

    // ---- write back: D layout  M = r + 8*(lane/16), N = lane%16 -----------
#pragma unroll
    for (int m = 0; m < 4; ++m)
#pragma unroll
        for (int n = 0; n < 2; ++n)
#pragma unroll
            for (int r = 0; r < 8; ++r) {
                const int gi = rowBase + wm + m * 16 + r + lh * 8;
                const int gj = colBase + wn + n * 16 + lr;
                gram[(size_t)gi * N_SAMP + gj] = acc[m][n][r];
            }
}

// ---------------------------------------------------------------------------
// Kernel 3: per-row hard mining -> rowloss[i] = relu(dist_ap - dist_an + m)
// ---------------------------------------------------------------------------
__global__ void tl_mine(const float* __restrict__ gram, const float* __restrict__ sq,
                        const int* __restrict__ tgt, float* __restrict__ rowloss) {
    __shared__ float rAp[256];
    __shared__ float rAn[256];
    const int i  = blockIdx.x;
    const int ti = tgt[i];
    const float sqi = sq[i];
    float ap = -INFINITY, an = INFINITY;
    for (int j = threadIdx.x; j < N_SAMP; j += 256) {
        const float g  = gram[(size_t)i * N_SAMP + j];
        const float d2 = sqi + sq[j] - 2.f * g;
        const float d  = sqrtf(fmaxf(d2, EPS_D));
        if (tgt[j] == ti) ap = fmaxf(ap, d);
        else              an = fminf(an, d);
    }
    rAp[threadIdx.x] = ap;
    rAn[threadIdx.x] = an;
    __syncthreads();
    for (int s = 128; s > 0; s >>= 1) {
        if (threadIdx.x < s) {
            rAp[threadIdx.x] = fmaxf(rAp[threadIdx.x], rAp[threadIdx.x + s]);
            rAn[threadIdx.x] = fminf(rAn[threadIdx.x], rAn[threadIdx.x + s]);
        }
        __syncthreads();
    }
    if (threadIdx.x == 0)
        rowloss[i] = fmaxf(0.f, rAp[0] - rAn[0] + MARGIN);
}

// ---------------------------------------------------------------------------
// Kernel 4: mean over rows -> scalar loss
// ---------------------------------------------------------------------------
__global__ void tl_mean(const float* __restrict__ rowloss, float* __restrict__ out) {
    __shared__ float red[256];
    float a = 0.f;
    for (int i = threadIdx.x; i < N_SAMP; i += 256) a += rowloss[i];
    red[threadIdx.x] = a;
    __syncthreads();
    for (int s = 128; s > 0; s >>= 1) {
        if (threadIdx.x < s) red[threadIdx.x] += red[threadIdx.x + s];
        __syncthreads();
    }
    if (threadIdx.x == 0) out[0] = red[0] / (float)N_SAMP;
}

extern "C" void kernel_launch(void* const* d_in, const int* in_sizes, int n_in,
                              void* d_out, int out_size, void* d_ws, size_t ws_size,
                              hipStream_t stream) {
    (void)in_sizes; (void)n_in; (void)out_size; (void)ws_size;
    const float* x   = (const float*)d_in[0];   // (1024, 8, 2048) f32
    const int*   tgt = (const int*)d_in[1];     // (1024,) labels
    const float* w   = (const float*)d_in[2];   // (2048,) f32, >= 0

    float* gram    = (float*)d_ws;                           // 1024*1024 f32 (4 MB)
    float* sq      = gram + (size_t)N_SAMP * N_SAMP;         // 1024 f32
    float* rowloss = sq + N_SAMP;                            // 1024 f32
    float* out     = (float*)d_out;

    tl_sqnorm<<<N_SAMP, 256, 0, stream>>>(x, w, sq);
    dim3 grid(N_SAMP / BN, N_SAMP / BM);
    tl_gram_wmma<<<grid, 256, 0, stream>>>(x, w, gram);
    tl_mine<<<N_SAMP, 256, 0, stream>>>(gram, sq, tgt, rowloss);
    tl_mean<<<1, 256, 0, stream>>>(rowloss, out);
}